// GeometricLMv2_62955630625403
// MI455X (gfx1250) — compile-verified
//
#include <hip/hip_runtime.h>
#include <hip/hip_bf16.h>
#include <math.h>

// Problem constants (from reference)
#define BQ 4096
#define SQ 32
#define DQ 256
#define VQ 50257
#define NQ 100   // N_ALIVE

typedef float v2f __attribute__((ext_vector_type(2)));
typedef float v8f __attribute__((ext_vector_type(8)));

// ---------------------------------------------------------------------------
// Kernel 1: one block per batch row b.
//   embeds = token_embed_w[token_ids[b]] + pos_embed_w        (staged in LDS)
//   attention-pool with context_query -> context[b, 0..255]
//   RBF over 100 positions -> normalized activations[b, 0..99] into d_ws
// ---------------------------------------------------------------------------
__global__ __launch_bounds__(256) void ctx_act_kernel(
    const int*   __restrict__ token_ids,     // [B,S]
    const float* __restrict__ token_embed_w, // [V,D]
    const float* __restrict__ pos_embed_w,   // [S,D]
    const float* __restrict__ context_query, // [D]
    const float* __restrict__ positions,     // [N,D]
    float*       __restrict__ act)           // [B,N] out (scratch)
{
    __shared__ float e[SQ][DQ + 1];   // +1 pad: kills LDS bank conflicts
    __shared__ float cqs[DQ];
    __shared__ int   toks[SQ];
    __shared__ float sw[SQ];          // scores, then softmax weights
    __shared__ float ctx[DQ];
    __shared__ float aN[NQ];
    __shared__ float asum;

    const int t    = threadIdx.x;     // 0..255 (== D)
    const int b    = blockIdx.x;
    const int lane = t & 31;
    const int wave = t >> 5;

    if (t < SQ) toks[t] = token_ids[b * SQ + t];
    cqs[t] = context_query[t];
    __syncthreads();

    // Build 32x256 embedding tile; each s-row is a fully coalesced 1KB load.
    for (int s = 0; s < SQ; ++s) {
        e[s][t] = token_embed_w[(size_t)toks[s] * DQ + t] + pos_embed_w[s * DQ + t];
    }
    __syncthreads();

    // Scores: 8 threads per s-row, reduce within 8-lane groups via shuffles.
    {
        const int s = t >> 3, dd = t & 7;
        float p = 0.f;
        for (int d = dd; d < DQ; d += 8) p += e[s][d] * cqs[d];
        p += __shfl_xor(p, 1, 32);
        p += __shfl_xor(p, 2, 32);
        p += __shfl_xor(p, 4, 32);
        if (dd == 0) sw[s] = p * 0.0625f;   // / sqrt(256)
    }
    __syncthreads();

    // Softmax over S=32 in wave 0 (one lane per s).
    if (wave == 0) {
        float sc = sw[lane];
        float mx = sc;
        for (int o = 16; o; o >>= 1) mx = fmaxf(mx, __shfl_xor(mx, o, 32));
        float ex = expf(sc - mx);
        float sm = ex;
        for (int o = 16; o; o >>= 1) sm += __shfl_xor(sm, o, 32);
        sw[lane] = ex / sm;
    }
    __syncthreads();

    // context[d] = sum_s w[s] * e[s][d]  (thread t == d; conflict-free via pad)
    {
        float c = 0.f;
        for (int s = 0; s < SQ; ++s) c = fmaf(sw[s], e[s][t], c);
        ctx[t] = c;
    }
    __syncthreads();

    // RBF: one lane per alive neuron; positions rows stream from L2.
    if (t < NQ) {
        const float* pr = positions + (size_t)t * DQ;
        float dsq = 0.f;
        for (int d = 0; d < DQ; ++d) {
            float df = ctx[d] - pr[d];
            dsq = fmaf(df, df, dsq);
        }
        // BANDWIDTH=0.5 -> exp(-dsq / (2*0.25)) = exp(-2*dsq)
        aN[t] = expf(-2.0f * dsq);
    }
    __syncthreads();

    if (wave == 0) {
        float p = 0.f;
        for (int n = lane; n < NQ; n += 32) p += aN[n];
        for (int o = 16; o; o >>= 1) p += __shfl_xor(p, o, 32);
        if (lane == 0) asum = p;
    }
    __syncthreads();

    if (t < NQ) act[(size_t)b * NQ + t] = aN[t] / (asum + 1e-8f);
}

// ---------------------------------------------------------------------------
// Kernel 2: logits[4096,50257] = act[4096,100] @ W[100,50257]
// fp32 WMMA (V_WMMA_F32_16X16X4_F32): K = 100 = 25 steps of 4, exact fp32.
// Block = 4 waves; each wave computes a 16x64 output strip (4 adjacent 16x16
// tiles reusing one A fragment). Store-bandwidth bound (824 MB out).
//
// Tail handling: column indices are CLAMPED to V-1 for loads (always legal;
// clamped lanes produce garbage that is never stored), so the inner K-loop is
// completely branch-free: 1x global_load_b64 (A) + 8x global_load_b32 (B) +
// 4x v_wmma per step. Store bounds check is a single wave-uniform branch.
// ---------------------------------------------------------------------------
__global__ __launch_bounds__(128) void rbf_gemm_kernel(
    const float* __restrict__ act,  // [B, NQ]
    const float* __restrict__ W,    // [NQ, V]
    float*       __restrict__ out)  // [B, V]
{
    const int lane  = threadIdx.x & 31;
    const int wave  = threadIdx.x >> 5;
    const int m0    = blockIdx.y * 16;
    const int ntile = blockIdx.x * 4 + wave;   // 64-wide strip index
    const int n0    = ntile * 64;
    if (n0 >= VQ) return;                      // wave-uniform: EXEC stays all-1s

    const int l15   = lane & 15;
    const int khalf = (lane >> 4) * 2;         // lanes 0-15 -> K{0,1}; 16-31 -> K{2,3}
    const float* arow = act + (size_t)(m0 + l15) * NQ + khalf;

    const int nA = n0 + 0  + l15;
    const int nB = n0 + 16 + l15;
    const int nC = n0 + 32 + l15;
    const int nD = n0 + 48 + l15;
    // Clamped load indices (branch-free; out-of-range lanes read V-1, unused).
    const int cA = nA < VQ ? nA : VQ - 1;
    const int cB = nB < VQ ? nB : VQ - 1;
    const int cC = nC < VQ ? nC : VQ - 1;
    const int cD = nD < VQ ? nD : VQ - 1;

    v8f acc0 = {}, acc1 = {}, acc2 = {}, acc3 = {};

#pragma unroll 5
    for (int kb = 0; kb < 25; ++kb) {
        const int k = kb * 4;
        v2f a;
        a.x = arow[k];       // K = k+khalf
        a.y = arow[k + 1];   // K = k+khalf+1

        const float* wr0 = W + (size_t)(k + khalf) * VQ;
        const float* wr1 = wr0 + VQ;

        v2f b0, b1, b2, b3;
        b0.x = wr0[cA]; b0.y = wr1[cA];
        b1.x = wr0[cB]; b1.y = wr1[cB];
        b2.x = wr0[cC]; b2.y = wr1[cC];
        b3.x = wr0[cD]; b3.y = wr1[cD];

        acc0 = __builtin_amdgcn_wmma_f32_16x16x4_f32(false, a, false, b0, (short)0, acc0, false, false);
        acc1 = __builtin_amdgcn_wmma_f32_16x16x4_f32(false, a, false, b1, (short)0, acc1, false, false);
        acc2 = __builtin_amdgcn_wmma_f32_16x16x4_f32(false, a, false, b2, (short)0, acc2, false, false);
        acc3 = __builtin_amdgcn_wmma_f32_16x16x4_f32(false, a, false, b3, (short)0, acc3, false, false);
    }

    // C/D layout: VGPR r -> row m0+r (lanes 0-15) and m0+8+r (lanes 16-31).
    const int rbase = m0 + (lane >> 4) * 8;
    if (n0 + 64 <= VQ) {
        // Full strip (785 of 786 strips): unguarded coalesced stores.
#pragma unroll
        for (int r = 0; r < 8; ++r) {
            const size_t ro = (size_t)(rbase + r) * VQ;
            out[ro + nA] = acc0[r];
            out[ro + nB] = acc1[r];
            out[ro + nC] = acc2[r];
            out[ro + nD] = acc3[r];
        }
    } else {
#pragma unroll
        for (int r = 0; r < 8; ++r) {
            const size_t ro = (size_t)(rbase + r) * VQ;
            if (nA < VQ) out[ro + nA] = acc0[r];
            if (nB < VQ) out[ro + nB] = acc1[r];
            if (nC < VQ) out[ro + nC] = acc2[r];
            if (nD < VQ) out[ro + nD] = acc3[r];
        }
    }
}

extern "C" void kernel_launch(void* const* d_in, const int* in_sizes, int n_in,
                              void* d_out, int out_size, void* d_ws, size_t ws_size,
                              hipStream_t stream) {
    const int*   token_ids     = (const int*)  d_in[0];  // [B,S]
    const float* token_embed_w = (const float*)d_in[1];  // [V,D]
    const float* pos_embed_w   = (const float*)d_in[2];  // [S,D]
    const float* context_query = (const float*)d_in[3];  // [D]
    const float* positions     = (const float*)d_in[4];  // [N,D]
    const float* output_w      = (const float*)d_in[5];  // [N,V]
    float* logits = (float*)d_out;                       // [B,V]

    float* act = (float*)d_ws;                           // [B,N] = 1.6 MB scratch

    ctx_act_kernel<<<dim3(BQ), dim3(256), 0, stream>>>(
        token_ids, token_embed_w, pos_embed_w, context_query, positions, act);

    const int ntiles64 = (VQ + 63) / 64;                 // 786
    dim3 grid((ntiles64 + 3) / 4, BQ / 16);              // (197, 256)
    rbf_gemm_kernel<<<grid, dim3(128), 0, stream>>>(act, output_w, logits);
}